// GraphClassifier_61280593379659
// MI455X (gfx1250) — compile-verified
//
#include <hip/hip_runtime.h>
#include <hip/hip_bf16.h>

#define NN 100000
#define EE 1600000
#define GG 1024
#define NFEAT 9
#define VOCAB 128
#define HH 128
#define LL 3
#define OUTD 10
#define BN_EPS 1e-5f

typedef float v2f __attribute__((ext_vector_type(2)));
typedef float v8f __attribute__((ext_vector_type(8)));

#define LDS_STRIDE 130  // pad transposed W rows to avoid 16-way LDS bank conflicts

// ---------------------------------------------------------------------------
// GEMM: C[nrows x 128] = A[nrows x 128] * W[128 x 128], f32 WMMA 16x16x4
// One wave computes a 16-row strip over all 128 columns (8 output tiles).
// ---------------------------------------------------------------------------
__global__ __launch_bounds__(256)
void gcn_gemm_wmma(const float* __restrict__ A, const float* __restrict__ W,
                   float* __restrict__ C, int nrows) {
  __shared__ float sWt[128 * LDS_STRIDE];  // W transposed: sWt[n*stride + k] = W[k][n]
  int tid = threadIdx.x;

  // cooperative load + transpose of W (128x128 f32 = 64KB)
  for (int i = tid; i < 128 * 32; i += 256) {
    int r = i >> 5;           // row of W (k)
    int c4 = (i & 31) << 2;   // col of W (n), 4 at a time
    const float4 w = *(const float4*)(W + r * 128 + c4);
    sWt[(c4 + 0) * LDS_STRIDE + r] = w.x;
    sWt[(c4 + 1) * LDS_STRIDE + r] = w.y;
    sWt[(c4 + 2) * LDS_STRIDE + r] = w.z;
    sWt[(c4 + 3) * LDS_STRIDE + r] = w.w;
  }
  __syncthreads();

  int wave = tid >> 5;
  int lane = tid & 31;
  int strip = blockIdx.x * 8 + wave;
  int row0 = strip * 16;
  if (row0 >= nrows) return;

  int m = lane & 15;             // M (for A) / N (for B) index within tile
  int kk = (lane >> 4) << 1;     // upper half-wave covers K+2,K+3

  v8f acc[8] = {};

  const float* arow = A + (size_t)(row0 + m) * HH;

  for (int k = 0; k < 128; k += 4) {
    // A fragment: 16x4 f32. VGPR0 = A[m][k+kk], VGPR1 = A[m][k+kk+1]
    v2f a = *(const v2f*)(arow + k + kk);
#pragma unroll
    for (int t = 0; t < 8; ++t) {
      // B fragment: 4x16 f32, column n = t*16+m, rows k+kk, k+kk+1
      v2f b = *(const v2f*)(&sWt[(t * 16 + m) * LDS_STRIDE + k + kk]);
      acc[t] = __builtin_amdgcn_wmma_f32_16x16x4_f32(
          /*neg_a=*/false, a, /*neg_b=*/false, b,
          /*c_mod=*/(short)0, acc[t], /*reuse_a=*/false, /*reuse_b=*/false);
    }
  }

  // D layout: VGPR v -> row (row0 + v + 8*(lane>>4)), col t*16 + (lane&15)
  int rbase = row0 + ((lane >> 4) << 3);
#pragma unroll
  for (int t = 0; t < 8; ++t) {
#pragma unroll
    for (int v = 0; v < 8; ++v) {
      C[(size_t)(rbase + v) * HH + t * 16 + m] = acc[t][v];
    }
  }
}

// ---------------------------------------------------------------------------
// Elementwise / scatter kernels
// ---------------------------------------------------------------------------
__global__ void gcn_fill_zero(float* __restrict__ p, int n) {
  int i = blockIdx.x * 256 + threadIdx.x;
  if (i < n) p[i] = 0.0f;
}

__global__ void gcn_degree(const int* __restrict__ dst, float* __restrict__ deg, int e) {
  int i = blockIdx.x * 256 + threadIdx.x;
  if (i < e) atomicAdd(&deg[dst[i]], 1.0f);
}

__global__ void gcn_dinv(float* __restrict__ d, int n) {
  int i = blockIdx.x * 256 + threadIdx.x;
  if (i < n) d[i] = rsqrtf(d[i] + 1.0f);
}

// AtomEncoder: h[n][c] = sum_f emb[f][x[n][f]][c]
__global__ __launch_bounds__(128)
void gcn_encoder(const int* __restrict__ x, const float* __restrict__ emb,
                 float* __restrict__ h) {
  int n = blockIdx.x;
  int c = threadIdx.x;
  float s = 0.0f;
#pragma unroll
  for (int f = 0; f < NFEAT; ++f) {
    int v = x[n * NFEAT + f];
    s += emb[((size_t)f * VOCAB + v) * HH + c];
  }
  h[(size_t)n * HH + c] = s;
}

// self-loop + bias init: hout = hw * dinv^2 + bias[c]
__global__ void gcn_selfinit(const float* __restrict__ hw, const float* __restrict__ dinv,
                             const float* __restrict__ bias, float* __restrict__ hout,
                             int total) {
  int i = blockIdx.x * 256 + threadIdx.x;
  if (i >= total) return;
  int n = i >> 7;
  int c = i & 127;
  float dv = dinv[n];
  hout[i] = hw[i] * dv * dv + bias[c];
}

// edge aggregation: one wave per edge, 4 channels per lane
__global__ void gcn_edgeagg(const int* __restrict__ src, const int* __restrict__ dst,
                            const float* __restrict__ dinv, const float* __restrict__ hw,
                            float* __restrict__ hout, int e) {
  int gid = blockIdx.x * 256 + threadIdx.x;
  int edge = gid >> 5;
  if (edge >= e) return;
  int q = (gid & 31) << 2;  // channel base
  int s = src[edge];
  int d = dst[edge];
  float ne = dinv[s] * dinv[d];
  const float4 v = *(const float4*)(hw + (size_t)s * HH + q);
  float* o = hout + (size_t)d * HH + q;
  atomicAdd(o + 0, v.x * ne);
  atomicAdd(o + 1, v.y * ne);
  atomicAdd(o + 2, v.z * ne);
  atomicAdd(o + 3, v.w * ne);
}

__global__ void gcn_bnrelu(float* __restrict__ h, const float* __restrict__ gamma,
                           const float* __restrict__ beta, const float* __restrict__ mean,
                           const float* __restrict__ var, int total) {
  int i = blockIdx.x * 256 + threadIdx.x;
  if (i >= total) return;
  int c = i & 127;
  float sc = gamma[c] * rsqrtf(var[c] + BN_EPS);
  float v = (h[i] - mean[c]) * sc + beta[c];
  h[i] = v > 0.0f ? v : 0.0f;
}

__global__ void gcn_poolcnt(const int* __restrict__ batch, float* __restrict__ counts, int n) {
  int i = blockIdx.x * 256 + threadIdx.x;
  if (i < n) atomicAdd(&counts[batch[i]], 1.0f);
}

__global__ void gcn_poolacc(const int* __restrict__ batch, const float* __restrict__ h,
                            float* __restrict__ pooled, int total) {
  int i = blockIdx.x * 256 + threadIdx.x;
  if (i >= total) return;
  int n = i >> 7;
  int c = i & 127;
  atomicAdd(&pooled[(size_t)batch[n] * HH + c], h[i]);
}

__global__ void gcn_classifier(const float* __restrict__ pooled, const float* __restrict__ counts,
                               const float* __restrict__ Wc, const float* __restrict__ bc,
                               float* __restrict__ out, int total) {
  int i = blockIdx.x * 256 + threadIdx.x;
  if (i >= total) return;
  int g = i / OUTD;
  int o = i % OUTD;
  float cnt = counts[g];
  cnt = cnt > 1.0f ? cnt : 1.0f;
  float inv = 1.0f / cnt;
  float s = bc[o];
#pragma unroll 4
  for (int c = 0; c < HH; ++c) {
    s += pooled[(size_t)g * HH + c] * inv * Wc[c * OUTD + o];
  }
  out[i] = s;
}

// ---------------------------------------------------------------------------
extern "C" void kernel_launch(void* const* d_in, const int* in_sizes, int n_in,
                              void* d_out, int out_size, void* d_ws, size_t ws_size,
                              hipStream_t stream) {
  (void)in_sizes; (void)n_in; (void)out_size; (void)ws_size;

  const int*   x        = (const int*)d_in[0];
  const int*   eidx     = (const int*)d_in[1];   // [2, E]
  const int*   batch    = (const int*)d_in[2];
  const float* emb      = (const float*)d_in[3];
  const float* Ws       = (const float*)d_in[4];
  const float* bs       = (const float*)d_in[5];
  const float* gamma    = (const float*)d_in[6];
  const float* beta     = (const float*)d_in[7];
  const float* run_mean = (const float*)d_in[8];
  const float* run_var  = (const float*)d_in[9];
  const float* W_clf    = (const float*)d_in[10];
  const float* b_clf    = (const float*)d_in[11];
  float* out = (float*)d_out;

  const int* src = eidx;
  const int* dst = eidx + EE;

  // workspace layout
  const size_t NH4 = (size_t)NN * HH * sizeof(float);  // 51.2 MB
  char* ws = (char*)d_ws;
  float* hA     = (float*)ws;                      // node features (current h)
  float* hB     = (float*)(ws + NH4);              // h @ W
  float* dinv   = (float*)(ws + 2 * NH4);          // N floats (also deg scratch)
  float* pooled = (float*)(ws + 2 * NH4 + (512u << 10));
  float* counts = (float*)(ws + 2 * NH4 + (512u << 10) + (size_t)GG * HH * sizeof(float));

  const int NH = NN * HH;           // 12,800,000
  const int nhBlocks = (NH + 255) / 256;

  // 1) degrees -> dinv
  gcn_fill_zero<<<(NN + 255) / 256, 256, 0, stream>>>(dinv, NN);
  gcn_degree<<<(EE + 255) / 256, 256, 0, stream>>>(dst, dinv, EE);
  gcn_dinv<<<(NN + 255) / 256, 256, 0, stream>>>(dinv, NN);

  // 2) atom encoder -> hA
  gcn_encoder<<<NN, 128, 0, stream>>>(x, emb, hA);

  // 3) GCN layers
  const int strips = NN / 16;                      // 6250
  const int gemmBlocks = (strips + 7) / 8;         // 782
  for (int l = 0; l < LL; ++l) {
    gcn_gemm_wmma<<<gemmBlocks, 256, 0, stream>>>(hA, Ws + (size_t)l * HH * HH, hB, NN);
    gcn_selfinit<<<nhBlocks, 256, 0, stream>>>(hB, dinv, bs + l * HH, hA, NH);
    {
      const long long tot = (long long)EE * 32;    // wave per edge
      gcn_edgeagg<<<(int)((tot + 255) / 256), 256, 0, stream>>>(src, dst, dinv, hB, hA, EE);
    }
    if (l < LL - 1) {
      gcn_bnrelu<<<nhBlocks, 256, 0, stream>>>(hA, gamma + l * HH, beta + l * HH,
                                               run_mean + l * HH, run_var + l * HH, NH);
    }
  }

  // 4) global mean pool
  gcn_fill_zero<<<(GG * HH + 255) / 256, 256, 0, stream>>>(pooled, GG * HH);
  gcn_fill_zero<<<(GG + 255) / 256, 256, 0, stream>>>(counts, GG);
  gcn_poolcnt<<<(NN + 255) / 256, 256, 0, stream>>>(batch, counts, NN);
  gcn_poolacc<<<nhBlocks, 256, 0, stream>>>(batch, hA, pooled, NH);

  // 5) classifier
  gcn_classifier<<<(GG * OUTD + 255) / 256, 256, 0, stream>>>(pooled, counts, W_clf, b_clf,
                                                              out, GG * OUTD);
}